// NormalEstimator_16149077033014
// MI455X (gfx1250) — compile-verified
//
#include <hip/hip_runtime.h>
#include <math.h>

// Reference shapes: depth [16,1,512,640] f32, out [16,3,512,640] f32.
constexpr int BATCH = 16;
constexpr int H = 512;
constexpr int W = 640;

constexpr int TW = 64;           // tile width (outputs)
constexpr int TH = 16;           // tile height (outputs)
constexpr int BY = 4;            // blockDim.y -> 64x4 = 256 threads = 8 wave32
constexpr int HALO_W = TW + 2;   // 66
constexpr int HALO_H = TH + 2;   // 18
constexpr int NTHREADS = TW * BY;

// ---- CDNA5 async global->LDS path (gfx1250), with safe fallback ----
#if defined(__AMDGCN__) && __has_builtin(__builtin_amdgcn_global_load_async_to_lds_b32)
#define USE_ASYNC_LDS 1
#else
#define USE_ASYNC_LDS 0
#endif

typedef __attribute__((address_space(1))) int* as1_int_p;
typedef __attribute__((address_space(3))) int* as3_int_p;

__device__ __forceinline__ as1_int_p to_as1(const void* p) {
    // generic global address == addrspace(1) address on amdgcn
    return (as1_int_p)(unsigned long long)p;
}
__device__ __forceinline__ as3_int_p to_as3(const void* p) {
    // generic LDS address carries the LDS byte offset in its low 32 bits
    return (as3_int_p)(unsigned int)(unsigned long long)p;
}

__device__ __forceinline__ void wait_async_zero() {
#if USE_ASYNC_LDS
#if __has_builtin(__builtin_amdgcn_s_wait_asynccnt)
    __builtin_amdgcn_s_wait_asynccnt(0);
#else
    asm volatile("s_wait_asynccnt 0" ::: "memory");
#endif
#endif
}

__global__ __launch_bounds__(NTHREADS) void normal_estimator_kernel(
    const float* __restrict__ depth,   // [BATCH,1,H,W]
    const float* __restrict__ vfov,    // [1]
    float* __restrict__ out)           // [BATCH,3,H,W]
{
    __shared__ float tile[HALO_H][HALO_W];

    const int b   = blockIdx.z;
    const int tx0 = blockIdx.x * TW;
    const int ty0 = blockIdx.y * TH;
    const int tid = threadIdx.y * TW + threadIdx.x;

    const float* __restrict__ dplane = depth + (size_t)b * H * W;

    // ---- cooperative halo load with replicate clamping (== 'edge' padding) ----
    for (int i = tid; i < HALO_H * HALO_W; i += NTHREADS) {
        const int ly = i / HALO_W;
        const int lx = i - ly * HALO_W;
        int gy = ty0 - 1 + ly; gy = gy < 0 ? 0 : (gy > H - 1 ? H - 1 : gy);
        int gx = tx0 - 1 + lx; gx = gx < 0 ? 0 : (gx > W - 1 ? W - 1 : gx);
#if USE_ASYNC_LDS
        __builtin_amdgcn_global_load_async_to_lds_b32(
            to_as1(dplane + (size_t)gy * W + gx), to_as3(&tile[ly][lx]),
            /*offset=*/0, /*cpol=*/0);
#else
        tile[ly][lx] = dplane[(size_t)gy * W + gx];
#endif
    }
    wait_async_zero();
    __syncthreads();

    // f computed in double to match Python's math.tan, then used as f32.
    const float vf = vfov[0];
    const float fs = (float)((double)H / (2.0 * tan((double)vf * 0.5)));

    const size_t plane = (size_t)H * W;
    float* __restrict__ out0 = out + (size_t)b * 3 * plane;

    #pragma unroll
    for (int k = 0; k < TH / BY; ++k) {
        const int r  = k * BY + threadIdx.y;     // row within tile
        const int ly = r + 1;
        const int lx = threadIdx.x + 1;

        const float tl = tile[ly - 1][lx - 1], tm = tile[ly - 1][lx], tr = tile[ly - 1][lx + 1];
        const float ml = tile[ly    ][lx - 1], cc = tile[ly    ][lx], mr = tile[ly    ][lx + 1];
        const float bl = tile[ly + 1][lx - 1], bm = tile[ly + 1][lx], br = tile[ly + 1][lx + 1];

        // cross-correlation with GX / GY (lax.conv does not flip kernels)
        const float gx = (tr - tl) + 2.0f * (mr - ml) + (br - bl);
        const float gy = (bl - tl) + 2.0f * (bm - tm) + (br - tr);

        const float s   = 8.0f * cc / fs;
        const float gxn = gx / s;
        const float gyn = gy / s;

        const float nx = -gxn;
        const float ny = -gyn;
        const float nrm = sqrtf(nx * nx + ny * ny + 1.0f);

        const bool invalid =
            (tl == 0.0f) | (tm == 0.0f) | (tr == 0.0f) |
            (ml == 0.0f) | (cc == 0.0f) | (mr == 0.0f) |
            (bl == 0.0f) | (bm == 0.0f) | (br == 0.0f);

        const float ox = invalid ? 0.0f : nx / nrm;
        const float oy = invalid ? 0.0f : ny / nrm;
        const float oz = invalid ? 0.0f : 1.0f / nrm;

        const size_t idx = (size_t)(ty0 + r) * W + (tx0 + threadIdx.x);
        out0[idx]              = ox;
        out0[plane + idx]      = oy;
        out0[2 * plane + idx]  = oz;
    }
}

extern "C" void kernel_launch(void* const* d_in, const int* in_sizes, int n_in,
                              void* d_out, int out_size, void* d_ws, size_t ws_size,
                              hipStream_t stream) {
    (void)in_sizes; (void)n_in; (void)out_size; (void)d_ws; (void)ws_size;
    const float* depth = (const float*)d_in[0];
    const float* vfov  = (const float*)d_in[1];
    float* out = (float*)d_out;

    dim3 block(TW, BY, 1);                     // 256 threads = 8 wave32
    dim3 grid(W / TW, H / TH, BATCH);          // 10 x 32 x 16 blocks
    normal_estimator_kernel<<<grid, block, 0, stream>>>(depth, vfov, out);
}